// TsallisBisect_12421045420947
// MI455X (gfx1250) — compile-verified
//
#include <hip/hip_runtime.h>
#include <cstdint>

#ifndef __has_builtin
#define __has_builtin(x) 0
#endif

#define AS1 __attribute__((address_space(1)))
#define AS3 __attribute__((address_space(3)))

typedef int v4i_vec __attribute__((vector_size(16)));   // b128 payload type per builtin signature

namespace {
constexpr int   kRows  = 4096;
constexpr int   kVocab = 32000;
constexpr int   kNV4   = kVocab / 4;      // 8000 float4 per row
constexpr int   kBlock = 1024;            // 32 wave32 waves, one WGP
constexpr int   kWaves = kBlock / 32;
constexpr int   kEptV4 = 8;               // 8*1024 = 8192 >= 8000 vec4 slots/thread
constexpr int   kIters = 50;
constexpr float kNegBig = -3.402823466e+38f;
constexpr float kTauGap = 0.005590169943749474f;  // (1/32000)^(alpha-1) = 1/sqrt(32000)
}

// ---------------- CDNA5 async global->LDS staging ----------------
__device__ __forceinline__ void async_g2l_b128(const float* g, float* l) {
#if __has_builtin(__builtin_amdgcn_global_load_async_to_lds_b128)
  __builtin_amdgcn_global_load_async_to_lds_b128(
      (AS1 v4i_vec*)(uintptr_t)g,
      (AS3 v4i_vec*)(unsigned)(uintptr_t)l,  // low 32 bits of a generic LDS pointer == LDS offset
      0, 0);
#else
  asm volatile("global_load_async_to_lds_b128 %0, %1, off"
               :: "v"((unsigned)(uintptr_t)l), "v"(g)
               : "memory");
#endif
}

__device__ __forceinline__ void wait_async0() {
#if __has_builtin(__builtin_amdgcn_s_wait_asynccnt)
  __builtin_amdgcn_s_wait_asynccnt(0);
#else
  asm volatile("s_wait_asynccnt 0" ::: "memory");
#endif
}

// ---------------- wave32 / block reductions ----------------
__device__ __forceinline__ float wave_max(float v) {
#pragma unroll
  for (int m = 16; m; m >>= 1) v = fmaxf(v, __shfl_xor(v, m, 32));
  return v;
}
__device__ __forceinline__ float wave_sum(float v) {
#pragma unroll
  for (int m = 16; m; m >>= 1) v += __shfl_xor(v, m, 32);
  return v;
}

__device__ __forceinline__ float block_max(float v, volatile float* red, volatile float* bc) {
  const int lane = threadIdx.x & 31, wave = threadIdx.x >> 5;
  v = wave_max(v);
  if (lane == 0) red[wave] = v;
  __syncthreads();
  if (wave == 0) {
    float t = red[lane];
    t = wave_max(t);
    if (lane == 0) *bc = t;
  }
  __syncthreads();
  return *bc;
}

__device__ __forceinline__ float block_sum(float v, volatile float* red, volatile float* bc) {
  const int lane = threadIdx.x & 31, wave = threadIdx.x >> 5;
  v = wave_sum(v);
  if (lane == 0) red[wave] = v;
  __syncthreads();
  if (wave == 0) {
    float t = red[lane];
    t = wave_sum(t);
    if (lane == 0) *bc = t;
  }
  __syncthreads();
  return *bc;
}

// ---------------- kernel: one workgroup per row ----------------
__global__ __launch_bounds__(kBlock)
void tsallis15_bisect_kernel(const float* __restrict__ X, float* __restrict__ out) {
  __shared__ __align__(16) float s_row[kNV4 * 4];  // 128 KB: staged row, then compacted active set
  __shared__ float s_red[kWaves];
  __shared__ float s_bc;
  __shared__ int   s_wbase[kWaves];
  __shared__ int   s_k;

  const int tid  = threadIdx.x;
  const int lane = tid & 31;
  const int wave = tid >> 5;
  const long long row = blockIdx.x;
  const float* __restrict__ grow = X + row * (long long)kVocab;
  float* __restrict__ orow = out + row * (long long)kVocab;

  // ---- Phase A: stage raw row global -> LDS via async b128 (ASYNCcnt) ----
#pragma unroll
  for (int i = 0; i < kEptV4; ++i) {
    const int s = i * kBlock + tid;
    if (s < kNV4) async_g2l_b128(grow + 4 * s, &s_row[4 * s]);
  }
  wait_async0();
  __syncthreads();

  // ---- Phase B: LDS -> registers (scaled by alpha-1 = 0.5), thread-local max ----
  float4 v[kEptV4];
  float lmax = kNegBig;
#pragma unroll
  for (int i = 0; i < kEptV4; ++i) {
    const int s = i * kBlock + tid;
    float4 t;
    if (s < kNV4) {
      t = *reinterpret_cast<const float4*>(&s_row[4 * s]);
      t.x *= 0.5f; t.y *= 0.5f; t.z *= 0.5f; t.w *= 0.5f;
    } else {
      t.x = t.y = t.z = t.w = kNegBig;   // pad: never active, never the max
    }
    v[i] = t;
    lmax = fmaxf(lmax, fmaxf(fmaxf(t.x, t.y), fmaxf(t.z, t.w)));
  }

  const float maxv  = block_max(lmax, s_red, &s_bc);
  const float tau0  = maxv - 1.0f;       // initial tau_lo; every tau_m >= tau0
  const float tauHi = maxv - kTauGap;

  // ---- Phase C: deterministic compaction of active set {xs > tau0} into LDS ----
  int n = 0;
#pragma unroll
  for (int i = 0; i < kEptV4; ++i)
    n += (v[i].x > tau0) + (v[i].y > tau0) + (v[i].z > tau0) + (v[i].w > tau0);

  int incl = n;                           // wave-level inclusive scan (deterministic order)
#pragma unroll
  for (int d = 1; d < 32; d <<= 1) {
    int t = __shfl_up(incl, d, 32);
    if (lane >= d) incl += t;
  }
  if (lane == 31) s_wbase[wave] = incl;   // per-wave totals
  __syncthreads();                        // also fences last raw reads of s_row
  if (wave == 0) {
    const int wt = s_wbase[lane];
    int wincl = wt;
#pragma unroll
    for (int d = 1; d < 32; d <<= 1) {
      int t = __shfl_up(wincl, d, 32);
      if (lane >= d) wincl += t;
    }
    s_wbase[lane] = wincl - wt;           // exclusive wave base
    if (lane == 31) s_k = wincl;          // total active count K (>=1: the max is active)
  }
  __syncthreads();

  int pos = s_wbase[wave] + (incl - n);
#pragma unroll
  for (int i = 0; i < kEptV4; ++i) {
    if (v[i].x > tau0) s_row[pos++] = v[i].x;
    if (v[i].y > tau0) s_row[pos++] = v[i].y;
    if (v[i].z > tau0) s_row[pos++] = v[i].z;
    if (v[i].w > tau0) s_row[pos++] = v[i].w;
  }
  __syncthreads();
  const int K = s_k;

  // Exact f over the compacted set: excluded elements contribute relu(<=0)^2 == 0.
  auto fsum = [&](float tau) -> float {
    float acc = 0.0f;
    for (int j = tid; j < K; j += kBlock) {
      const float t = fmaxf(s_row[j] - tau, 0.0f);
      acc = fmaf(t, t, acc);
    }
    return block_sum(acc, s_red, &s_bc);
  };

  // ---- Phase D: bisection (all scalar state uniform across the block) ----
  const float f_lo = fsum(tau0) - 1.0f;
  float tau_lo = tau0;
  float dm     = tauHi - tau0;
  float tau_m  = tau0;
#pragma unroll 1
  for (int it = 0; it < kIters; ++it) {
    dm *= 0.5f;
    tau_m = tau_lo + dm;
    const float f_m = fsum(tau_m) - 1.0f;
    if (f_m * f_lo >= 0.0f) tau_lo = tau_m;
  }

  // ---- Phase E: normalize and stream out (single full pass from registers) ----
  const float Z   = fsum(tau_m);          // sum of p_m, exact over compacted set
  const float inv = 1.0f / Z;
#pragma unroll
  for (int i = 0; i < kEptV4; ++i) {
    const int s = i * kBlock + tid;
    if (s < kNV4) {
      const float4 t = v[i];
      float4 r; float a;
      a = fmaxf(t.x - tau_m, 0.0f); r.x = a * a * inv;
      a = fmaxf(t.y - tau_m, 0.0f); r.y = a * a * inv;
      a = fmaxf(t.z - tau_m, 0.0f); r.z = a * a * inv;
      a = fmaxf(t.w - tau_m, 0.0f); r.w = a * a * inv;
      *reinterpret_cast<float4*>(&orow[4 * s]) = r;
    }
  }
}

extern "C" void kernel_launch(void* const* d_in, const int* in_sizes, int n_in,
                              void* d_out, int out_size, void* d_ws, size_t ws_size,
                              hipStream_t stream) {
  (void)in_sizes; (void)n_in; (void)d_ws; (void)ws_size; (void)out_size;
  const float* X = (const float*)d_in[0];
  float* out     = (float*)d_out;
  hipLaunchKernelGGL(tsallis15_bisect_kernel, dim3(kRows), dim3(kBlock), 0, stream, X, out);
}